// TransNetImplicit_38259568672787
// MI455X (gfx1250) — compile-verified
//
#include <hip/hip_runtime.h>
#include <hip/hip_bf16.h>
#include <math.h>

// ---------------------------------------------------------------------------
// TransNet implicit layers on MI455X (gfx1250).
//   v0=(x@W_in+b_in)^T; 8x { solve [[I,B],[-B^T,2I]] [z0;z1] = [v0+.1q ; v1+tanh(v0)] }
//   out = v0^T @ W_out + b_out
// Schur reduction: z1 = [Minv*B^T | Minv] @ [r0;r1],  z0 = r0 - B@z1,
//   Minv = (2I + B^T B)^-1  (SPD, well conditioned -> pivot-free Gauss-Jordan).
// Activations (256 x 64 batch tile) live in LDS across all 8 layers; all GEMMs
// use V_WMMA_F32_16X16X4_F32 (fp32 matrix path); x / W_out staged into LDS via
// gfx1250 async-to-LDS loads tracked by ASYNCcnt.
// ---------------------------------------------------------------------------

typedef __attribute__((ext_vector_type(2))) float v2f;
typedef __attribute__((ext_vector_type(8))) float v8f;

#define U       256
#define TN      64          // batch columns per workgroup
#define KIN     784
#define NOUT    10
#define NLAYERS 8
#define KC      56          // x K-chunk staged in LDS (784 = 14*56)
#define NCH     14

// LDS partition (floats), dynamic LDS assumed to start at LDS offset 0
// (no static __shared__ in this kernel).
#define RT_F    (TN * 512)                 // R^T  : [n][k] k<256 = r0/v0, k>=256 = r1
#define ZT_F    (TN * 256)                 // Z^T  : [n][m] = z1/v1
#define XS_F    (TN * KC)                  // x staging chunk, n-major
#define WS_F    (U * NOUT)                 // W_out staged
#define XS_OFF_B  ((RT_F + ZT_F) * 4)                // 196608
#define WS_OFF_B  ((RT_F + ZT_F + XS_F) * 4)         // 210944
#define LDS_BYTES ((RT_F + ZT_F + XS_F + WS_F) * 4)  // 221184

__device__ __forceinline__ v8f wmma4(v2f a, v2f b, v8f c) {
  // D = A(16x4 f32) * B(4x16 f32) + C(16x16 f32)
  return __builtin_amdgcn_wmma_f32_16x16x4_f32(
      /*neg_a=*/false, a, /*neg_b=*/false, b,
      /*c_mod=*/(short)0, c, /*reuse_a=*/false, /*reuse_b=*/false);
}

__device__ __forceinline__ void async_b128(unsigned lds_off, const float* g) {
  asm volatile("global_load_async_to_lds_b128 %0, %1, off"
               :: "v"(lds_off), "v"(g) : "memory");
}
__device__ __forceinline__ void wait_async0() {
  asm volatile("s_wait_asynccnt 0x0" ::: "memory");
}

// ---------------------------------------------------------------------------
// Prep 1: W_in (784x256) -> W_inT (256x784) row-major for WMMA A operand.
__global__ void prep_transpose(const float* __restrict__ Win,
                               float* __restrict__ WinT) {
  int k = blockIdx.x;        // 0..783
  int m = threadIdx.x;       // 0..255
  WinT[(size_t)m * KIN + k] = Win[(size_t)k * U + m];
}

// ---------------------------------------------------------------------------
// Prep 2: per layer, C_i = [Minv * B^T | Minv] (256x512), Minv = (2I+B^T B)^-1.
// One workgroup (1024 thr, 256KB LDS) per layer.
__global__ __launch_bounds__(1024) void prep_layer(const float* __restrict__ B0,
                                                   float* __restrict__ Cmat) {
  extern __shared__ float M[];                    // 256*256 fp32
  const int layer = blockIdx.x;
  const float* Bm = B0 + (size_t)layer * U * U;   // row-major 256x256
  const int tid = threadIdx.x;

  // (a) M = 2I + B^T B
  for (int e = tid; e < U * U; e += 1024) {
    int r = e >> 8, c = e & 255;
    float s = (r == c) ? 2.0f : 0.0f;
    for (int k = 0; k < U; ++k) s += Bm[k * U + r] * Bm[k * U + c];
    M[e] = s;
  }
  __syncthreads();

  // (b) in-place pivot-free Gauss-Jordan inversion of M
  const int row = tid >> 2;            // 0..255
  const int cq  = (tid & 3) * 64;      // 64-col quarter
  for (int k = 0; k < U; ++k) {
    float p = 1.0f / M[k * U + k];
    float f = M[row * U + k];
    __syncthreads();
    if (row != k) {
      float fp = f * p;
      for (int c = cq; c < cq + 64; ++c)
        if (c != k) M[row * U + c] -= fp * M[k * U + c];
      if (k >= cq && k < cq + 64) M[row * U + k] = -fp;
    }
    __syncthreads();
    if (row == k) {
      for (int c = cq; c < cq + 64; ++c)
        M[k * U + c] = (c == k) ? p : M[k * U + c] * p;
    }
    __syncthreads();
  }

  // (c) C_i[:,0:256] = Minv @ B^T ; C_i[:,256:512] = Minv
  float* Ci = Cmat + (size_t)layer * U * 512;
  for (int e = tid; e < U * U; e += 1024) {
    int r = e >> 8, c = e & 255;
    float s = 0.0f;
    for (int k = 0; k < U; ++k) s += M[r * U + k] * Bm[c * U + k];
    Ci[(size_t)r * 512 + c]       = s;
    Ci[(size_t)r * 512 + 256 + c] = M[r * U + c];
  }
}

// ---------------------------------------------------------------------------
// Main fused kernel: one workgroup owns 64 batch columns end-to-end.
__global__ __launch_bounds__(256) void transnet_main(
    const float* __restrict__ x,     const float* __restrict__ WinT,
    const float* __restrict__ b_in,  const float* __restrict__ B0,
    const float* __restrict__ q,     const float* __restrict__ Cmat,
    const float* __restrict__ Wout,  const float* __restrict__ bout,
    float* __restrict__ out) {
  extern __shared__ float lds[];
  float* Rt = lds;                // [TN][512]
  float* Zt = Rt + RT_F;          // [TN][256]
  float* Xs = Zt + ZT_F;          // [TN][KC]
  float* Ws = Xs + XS_F;          // [256][10]

  const int tid  = threadIdx.x;
  const int lane = tid & 31;
  const int wave = tid >> 5;          // 0..7
  const int hl   = lane >> 4;         // lane half (K sub-select)
  const int lm   = lane & 15;         // M/N within tile
  const int m_base = wave * 32;       // each wave owns a 32-row output strip
  const int n0 = blockIdx.x * TN;     // batch tile start

  // Stage W_out (256x10 = 640 x b128) asynchronously up front.
  for (int j = tid; j < WS_F / 4; j += 256)
    async_b128((unsigned)(WS_OFF_B + j * 16), Wout + j * 4);

  // ---- Input GEMM: v0 = W_in^T @ x^T + b_in (M=256, N=64, K=784) ----
  v8f acc[2][4] = {};
  for (int ch = 0; ch < NCH; ++ch) {
    // async-stage x[:, ch*56 : ch*56+56] for our 64 batch rows (n-major)
    for (int j = tid; j < TN * (KC / 4); j += 256) {
      int n = j / (KC / 4), t = j % (KC / 4);
      async_b128((unsigned)(XS_OFF_B + n * (KC * 4) + t * 16),
                 x + (size_t)(n0 + n) * KIN + ch * KC + t * 4);
    }
    wait_async0();
    __syncthreads();
#pragma unroll 2
    for (int kk = 0; kk < KC; kk += 4) {
      int kg = ch * KC + kk + 2 * hl;
      v2f a0 = *(const v2f*)(WinT + (size_t)(m_base + lm) * KIN + kg);
      v2f a1 = *(const v2f*)(WinT + (size_t)(m_base + 16 + lm) * KIN + kg);
#pragma unroll
      for (int nt = 0; nt < 4; ++nt) {
        v2f b = *(const v2f*)(Xs + (nt * 16 + lm) * KC + kk + 2 * hl);
        acc[0][nt] = wmma4(a0, b, acc[0][nt]);
        acc[1][nt] = wmma4(a1, b, acc[1][nt]);
      }
    }
    __syncthreads();
  }
  // v0 = acc + b_in -> Rt[:,0:256]; v1 = tanh(v0) -> Zt
#pragma unroll
  for (int mt = 0; mt < 2; ++mt)
#pragma unroll
    for (int nt = 0; nt < 4; ++nt)
#pragma unroll
      for (int v = 0; v < 8; ++v) {
        int m = m_base + mt * 16 + v + 8 * hl;
        int n = nt * 16 + lm;
        float val = acc[mt][nt][v] + b_in[m];
        Rt[n * 512 + m] = val;
        Zt[n * 256 + m] = tanhf(val);
      }
  __syncthreads();

  // ---- 8 implicit layers, fully LDS-resident ----
  for (int layer = 0; layer < NLAYERS; ++layer) {
    // elementwise: r1 = v1 + tanh(v0) ; r0 = v0 + 0.1*q
    const float* qi = q + layer * U;
    for (int idx = tid; idx < TN * U; idx += 256) {
      int n = idx >> 8, m = idx & 255;
      float v0v = Rt[n * 512 + m];
      Rt[n * 512 + 256 + m] = Zt[n * 256 + m] + tanhf(v0v);
      Rt[n * 512 + m] = v0v + 0.1f * qi[m];
    }
    __syncthreads();

    // GEMM1: z1 = C_i(256x512) @ R(512xTN) -> Zt
    {
      const float* Ci = Cmat + (size_t)layer * U * 512;
      v8f za[2][4] = {};
#pragma unroll 2
      for (int k = 0; k < 512; k += 4) {
        int kg = k + 2 * hl;
        v2f a0 = *(const v2f*)(Ci + (size_t)(m_base + lm) * 512 + kg);
        v2f a1 = *(const v2f*)(Ci + (size_t)(m_base + 16 + lm) * 512 + kg);
#pragma unroll
        for (int nt = 0; nt < 4; ++nt) {
          v2f b = *(const v2f*)(Rt + (nt * 16 + lm) * 512 + kg);
          za[0][nt] = wmma4(a0, b, za[0][nt]);
          za[1][nt] = wmma4(a1, b, za[1][nt]);
        }
      }
#pragma unroll
      for (int mt = 0; mt < 2; ++mt)
#pragma unroll
        for (int nt = 0; nt < 4; ++nt)
#pragma unroll
          for (int v = 0; v < 8; ++v)
            Zt[(nt * 16 + lm) * 256 + m_base + mt * 16 + v + 8 * hl] =
                za[mt][nt][v];
    }
    __syncthreads();

    // GEMM2: v0_new = r0 - B_i(256x256) @ z1(256xTN), in place in Rt[:,0:256]
    {
      const float* Bi = B0 + (size_t)layer * U * U;
      v8f zb[2][4] = {};
#pragma unroll 2
      for (int k = 0; k < 256; k += 4) {
        int kg = k + 2 * hl;
        v2f a0 = *(const v2f*)(Bi + (size_t)(m_base + lm) * U + kg);
        v2f a1 = *(const v2f*)(Bi + (size_t)(m_base + 16 + lm) * U + kg);
#pragma unroll
        for (int nt = 0; nt < 4; ++nt) {
          v2f b = *(const v2f*)(Zt + (nt * 16 + lm) * 256 + kg);
          zb[0][nt] = wmma4(a0, b, zb[0][nt]);
          zb[1][nt] = wmma4(a1, b, zb[1][nt]);
        }
      }
#pragma unroll
      for (int mt = 0; mt < 2; ++mt)
#pragma unroll
        for (int nt = 0; nt < 4; ++nt)
#pragma unroll
          for (int v = 0; v < 8; ++v) {
            int m = m_base + mt * 16 + v + 8 * hl;
            int n = nt * 16 + lm;
            Rt[n * 512 + m] -= zb[mt][nt][v];
          }
    }
    __syncthreads();
  }

  // ---- Output: out = v0^T @ W_out + b_out (TN x 10) ----
  for (int j = tid; j < TN * NOUT; j += 256) {
    int n = j / NOUT, o = j % NOUT;
    float s = bout[o];
    for (int m = 0; m < U; ++m) s += Rt[n * 512 + m] * Ws[m * NOUT + o];
    out[(size_t)(n0 + n) * NOUT + o] = s;
  }
}

// ---------------------------------------------------------------------------
extern "C" void kernel_launch(void* const* d_in, const int* in_sizes, int n_in,
                              void* d_out, int out_size, void* d_ws, size_t ws_size,
                              hipStream_t stream) {
  const float* x    = (const float*)d_in[0];
  const float* Win  = (const float*)d_in[1];
  const float* b_in = (const float*)d_in[2];
  const float* B0   = (const float*)d_in[3];
  const float* q    = (const float*)d_in[4];
  const float* Wout = (const float*)d_in[5];
  const float* bout = (const float*)d_in[6];
  float* out = (float*)d_out;

  float* ws   = (float*)d_ws;
  float* WinT = ws;                         // 256*784 floats
  float* Cmat = ws + (size_t)U * KIN;       // 8*256*512 floats (~4.8 MB total)

  prep_transpose<<<KIN, U, 0, stream>>>(Win, WinT);
  prep_layer<<<NLAYERS, 1024, U * U * sizeof(float), stream>>>(B0, Cmat);
  transnet_main<<<32768 / TN, 256, LDS_BYTES, stream>>>(
      x, WinT, b_in, B0, q, Cmat, Wout, bout, out);
}